// AFSDRefineHead_37649683316932
// MI455X (gfx1250) — compile-verified
//
#include <hip/hip_runtime.h>
#include <hip/hip_bf16.h>

typedef float v2f __attribute__((ext_vector_type(2)));
typedef float v8f __attribute__((ext_vector_type(8)));

#define NLVL 128   // T_lvl == N proposals
#define BATCH 2

// ---------------------------------------------------------------------------
// GEMM via V_WMMA_F32_16X16X4_F32:  Y[b,m,n] = sum_k W[m,k]*X[b,k,n] + bias[m]
// W: [M,K] row-major; X: [B,K,N]; Y: [B,M,N].  M%64==0, K%32==0, N%16==0.
// N is a compile-time constant so every B-fragment load folds its k-offset
// into a VMEM immediate (no per-iteration 64-bit address VALU).
// One wave -> one 16x16 tile of D; 4 waves per block.
// A layout (f32 16x4): lanes 0-15 rows M=0..15 K={0,1}; lanes 16-31 K={2,3}.
// B layout (f32 4x16): VGPR v: lanes 0-15 -> K=v, lanes 16-31 -> K=v+2.
// D layout: VGPR v: lanes 0-15 -> M=v, lanes 16-31 -> M=v+8, N=lane&15.
// ---------------------------------------------------------------------------
template <int N>
__global__ __launch_bounds__(128) void gemm_wmma_f32(
    const float* __restrict__ W, const float* __restrict__ X,
    const float* __restrict__ bias, float* __restrict__ Y,
    int M, int K)
{
  const int lane = threadIdx.x;                       // 0..31
  const int m0   = (blockIdx.x * 4 + threadIdx.y) * 16;
  const int n0   = blockIdx.y * 16;
  const int b    = blockIdx.z;
  const int hi   = lane >> 4;                         // 0 or 1
  const int l15  = lane & 15;

  const int arow = m0 + l15;
  const int bcol = n0 + l15;
  const float* __restrict__ wptr = W + (size_t)arow * K + 2 * hi;
  const float* __restrict__ xptr = X + (size_t)b * K * N + (size_t)(2 * hi) * N + bcol;

  v8f acc = {};
  // 8 WMMAs (K-chunk of 32) per iteration; all loads use immediate offsets.
#pragma unroll 1
  for (int k = 0; k < K; k += 32) {
#pragma unroll
    for (int j = 0; j < 8; ++j) {
      v2f a = *(const v2f*)(wptr + 4 * j);            // 8B aligned (K%4==0, even)
      v2f bv;
      bv.x = xptr[4 * j * N];
      bv.y = xptr[(4 * j + 1) * N];
      acc = __builtin_amdgcn_wmma_f32_16x16x4_f32(
          /*neg_a=*/false, a, /*neg_b=*/false, bv,
          /*c_mod=*/(short)0, acc, /*reuse_a=*/false, /*reuse_b=*/false);
    }
    wptr += 32;
    xptr += 32 * N;
  }

  float* __restrict__ yb = Y + (size_t)b * M * N;
#pragma unroll
  for (int v = 0; v < 8; ++v) {
    const int m = m0 + v + 8 * hi;
    yb[(size_t)m * N + bcol] = acc[v] + bias[m];
  }
}

// ---------------------------------------------------------------------------
// GroupNorm(32 groups) + ReLU.  In: Y [B,C,N]. Out written at channel offset
// chOff into a buffer with outC channels (enables fused concat placement).
// One block per (group, batch); biased variance (matches jnp.var); eps=1e-5.
// ---------------------------------------------------------------------------
__global__ __launch_bounds__(256) void gn_relu(
    const float* __restrict__ Y, const float* __restrict__ gamma,
    const float* __restrict__ beta, float* __restrict__ out,
    int C, int N, int outC, int chOff)
{
  const int g  = blockIdx.x;          // 0..31
  const int b  = blockIdx.y;
  const int gs = C >> 5;              // channels per group
  const int cnt = gs * N;
  const float* __restrict__ yg = Y + ((size_t)b * C + (size_t)g * gs) * N;

  float s = 0.f, sq = 0.f;
  for (int i = threadIdx.x; i < cnt; i += 256) {
    float v = yg[i];
    s  += v;
    sq += v * v;
  }
  __shared__ float ss[256], sv[256];
  ss[threadIdx.x] = s; sv[threadIdx.x] = sq;
  __syncthreads();
  for (int off = 128; off > 0; off >>= 1) {
    if (threadIdx.x < off) {
      ss[threadIdx.x] += ss[threadIdx.x + off];
      sv[threadIdx.x] += sv[threadIdx.x + off];
    }
    __syncthreads();
  }
  const float inv_n = 1.0f / (float)cnt;
  const float mu    = ss[0] * inv_n;
  const float var   = sv[0] * inv_n - mu * mu;
  const float rstd  = rsqrtf(var + 1e-5f);

  float* __restrict__ og = out + ((size_t)b * outC + chOff + (size_t)g * gs) * N;
  for (int i = threadIdx.x; i < cnt; i += 256) {
    const int c = g * gs + i / N;
    float v = (yg[i] - mu) * rstd * gamma[c] + beta[c];
    og[i] = v > 0.f ? v : 0.f;
  }
}

// ---------------------------------------------------------------------------
// Boundary max pooling: feat [B,C,T], seg [B,N,4] (ls,le,rs,re) inclusive.
// ch<half uses [ls,le], else [rs,re].  Out written at chOff into outC buffer.
// ---------------------------------------------------------------------------
__global__ void boundary_pool(
    const float* __restrict__ feat, const int* __restrict__ seg,
    float* __restrict__ out, int C, int T, int N, int half, int outC, int chOff)
{
  const int idx = blockIdx.x * blockDim.x + threadIdx.x;
  const int total = BATCH * C * N;
  if (idx >= total) return;
  const int n  = idx % N;
  const int ch = (idx / N) % C;
  const int b  = idx / (N * C);

  const int* sg = seg + ((size_t)b * N + n) * 4;
  int s, e;
  if (ch < half) { s = sg[0]; e = sg[1]; } else { s = sg[2]; e = sg[3]; }
  s = s < 0 ? 0 : s;
  e = e > T - 1 ? T - 1 : e;

  const float* __restrict__ f = feat + ((size_t)b * C + ch) * T;
  float m = -INFINITY;
  for (int t = s; t <= e; ++t) m = fmaxf(m, f[t]);
  out[((size_t)b * outC + chOff + ch) * N + n] = m;
}

// ---------------------------------------------------------------------------
// out[b,t,c] = in[b, cOff+c, t]   (channel-slice transpose)  out: [B,T,cCnt]
// ---------------------------------------------------------------------------
__global__ void transpose_slice(
    const float* __restrict__ in, float* __restrict__ out,
    int Cin, int T, int cOff, int cCnt)
{
  const int idx = blockIdx.x * blockDim.x + threadIdx.x;
  const int total = BATCH * T * cCnt;
  if (idx >= total) return;
  const int c = idx % cCnt;
  const int t = (idx / cCnt) % T;
  const int b = idx / (cCnt * T);
  out[idx] = in[((size_t)b * Cin + cOff + c) * T + t];
}

// out[b,n,o] = sum_c W[o,c]*X[b,c,n] + bias[o]   (tiny head, O in {2,21})
__global__ void head_gemm(
    const float* __restrict__ W, const float* __restrict__ bias,
    const float* __restrict__ X, float* __restrict__ out, int C, int N, int O)
{
  const int idx = blockIdx.x * blockDim.x + threadIdx.x;
  const int total = BATCH * N * O;
  if (idx >= total) return;
  const int o = idx % O;
  const int n = (idx / O) % N;
  const int b = idx / (O * N);
  float acc = bias[o];
  const float* __restrict__ w  = W + (size_t)o * C;
  const float* __restrict__ xb = X + (size_t)b * C * N + n;
  for (int c = 0; c < C; ++c) acc += w[c] * xb[(size_t)c * N];
  out[idx] = acc;
}

// 1-out-channel 3-tap 'SAME' conv over time: out[b,n] = b0 + sum_{c,k} w[c,k]*X[b,c,n+k-1]
__global__ void center_conv(
    const float* __restrict__ w, const float* __restrict__ bias,
    const float* __restrict__ X, float* __restrict__ out, int C, int N)
{
  const int idx = blockIdx.x * blockDim.x + threadIdx.x;
  if (idx >= BATCH * N) return;
  const int n = idx % N;
  const int b = idx / N;
  float acc = bias[0];
  const float* __restrict__ xb = X + (size_t)b * C * N;
  for (int c = 0; c < C; ++c) {
    const float* wc = w + 3 * c;
    const float* xc = xb + (size_t)c * N;
    if (n - 1 >= 0) acc += wc[0] * xc[n - 1];
    acc += wc[1] * xc[n];
    if (n + 1 < N) acc += wc[2] * xc[n + 1];
  }
  out[idx] = acc;
}

// ---------------------------------------------------------------------------
// Host-side orchestration
// ---------------------------------------------------------------------------
struct Branch {
  const float *cur_b, *cur_be, *cur_g, *cur_w;
  const float *lr_b, *lr_be, *lr_g, *lr_w;
  const float *prop_b, *prop_be, *prop_g, *prop_w;
  const float *roi_b, *roi_be, *roi_g, *roi_w;
};

static Branch get_branch(void* const* d_in, int base) {
  Branch br;
  br.cur_b  = (const float*)d_in[base + 0];
  br.cur_be = (const float*)d_in[base + 1];
  br.cur_g  = (const float*)d_in[base + 2];
  br.cur_w  = (const float*)d_in[base + 3];
  br.lr_b   = (const float*)d_in[base + 4];
  br.lr_be  = (const float*)d_in[base + 5];
  br.lr_g   = (const float*)d_in[base + 6];
  br.lr_w   = (const float*)d_in[base + 7];
  br.prop_b = (const float*)d_in[base + 8];
  br.prop_be= (const float*)d_in[base + 9];
  br.prop_g = (const float*)d_in[base + 10];
  br.prop_w = (const float*)d_in[base + 11];
  br.roi_b  = (const float*)d_in[base + 12];
  br.roi_be = (const float*)d_in[base + 13];
  br.roi_g  = (const float*)d_in[base + 14];
  br.roi_w  = (const float*)d_in[base + 15];
  return br;
}

static void run_branch(hipStream_t stream, const Branch& br,
                       const float* feat, const float* flf,
                       const int* segments, const int* frame_segments,
                       float* y1024, float* feat2, float* y512,
                       float* pool512, float* cat, float* propO)
{
  const dim3 wv(32, 4);
  // cur: [512x512] @ feat -> GN+ReLU -> cat rows [1536,2048)
  gemm_wmma_f32<NLVL><<<dim3(8, 8, BATCH), wv, 0, stream>>>(br.cur_w, feat, br.cur_b, y512, 512, 512);
  gn_relu<<<dim3(32, BATCH), 256, 0, stream>>>(y512, br.cur_g, br.cur_be, cat, 512, NLVL, 2048, 1536);
  // lr: [1024x512] @ feat -> GN+ReLU -> feat2
  gemm_wmma_f32<NLVL><<<dim3(16, 8, BATCH), wv, 0, stream>>>(br.lr_w, feat, br.lr_b, y1024, 1024, 512);
  gn_relu<<<dim3(32, BATCH), 256, 0, stream>>>(y1024, br.lr_g, br.lr_be, feat2, 1024, NLVL, 1024, 0);
  // boundary pool feat2 -> cat rows [512,1536)
  {
    int total = BATCH * 1024 * NLVL;
    boundary_pool<<<(total + 255) / 256, 256, 0, stream>>>(feat2, segments, cat, 1024, NLVL, NLVL, 512, 2048, 512);
  }
  // boundary pool frame-level feature -> pool512
  {
    int total = BATCH * 512 * NLVL;
    boundary_pool<<<(total + 255) / 256, 256, 0, stream>>>(flf, frame_segments, pool512, 512, 512, NLVL, 256, 512, 0);
  }
  // roi: [512x512] @ pool512 -> GN+ReLU -> cat rows [0,512)
  gemm_wmma_f32<NLVL><<<dim3(8, 8, BATCH), wv, 0, stream>>>(br.roi_w, pool512, br.roi_b, y512, 512, 512);
  gn_relu<<<dim3(32, BATCH), 256, 0, stream>>>(y512, br.roi_g, br.roi_be, cat, 512, NLVL, 2048, 0);
  // prop: [512x2048] @ cat -> GN+ReLU -> propO
  gemm_wmma_f32<NLVL><<<dim3(8, 8, BATCH), wv, 0, stream>>>(br.prop_w, cat, br.prop_b, y512, 512, 2048);
  gn_relu<<<dim3(32, BATCH), 256, 0, stream>>>(y512, br.prop_g, br.prop_be, propO, 512, NLVL, 512, 0);
}

extern "C" void kernel_launch(void* const* d_in, const int* in_sizes, int n_in,
                              void* d_out, int out_size, void* d_ws, size_t ws_size,
                              hipStream_t stream) {
  // Positional inputs
  const float* flf        = (const float*)d_in[0];  // [2,512,512]
  const float* loc_feat   = (const float*)d_in[1];  // [2,512,128]
  const float* conf_feat  = (const float*)d_in[2];  // [2,512,128]
  const int*   segments   = (const int*)d_in[3];    // [2,128,4]
  const int*   frame_seg  = (const int*)d_in[4];    // [2,128,4]
  // params pytree, leaves in sorted-key order:
  // 5:center_b 6:center_w 7..22:conf{...} 23:conf_head_b 24:conf_head_w
  // 25..40:loc{...} 41:loc_head_b 42:loc_head_w
  const float* center_b    = (const float*)d_in[5];
  const float* center_w    = (const float*)d_in[6];
  const Branch conf        = get_branch(d_in, 7);
  const float* conf_head_b = (const float*)d_in[23];
  const float* conf_head_w = (const float*)d_in[24];
  const Branch loc         = get_branch(d_in, 25);
  const float* loc_head_b  = (const float*)d_in[41];
  const float* loc_head_w  = (const float*)d_in[42];

  // Workspace (floats), reused sequentially across branches (~5.8 MB)
  float* ws      = (float*)d_ws;
  float* y1024   = ws;                  // 2*1024*128 = 262144
  float* feat2   = y1024  + 262144;     // 262144
  float* y512    = feat2  + 262144;     // 2*512*128 = 131072
  float* pool512 = y512   + 131072;     // 131072
  float* cat     = pool512 + 131072;    // 2*2048*128 = 524288
  float* propO   = cat    + 524288;     // 131072

  // Output slices (flat, return order)
  float* out         = (float*)d_out;
  float* o_start     = out;             // [2,512,256]
  float* o_end       = out + 262144;    // [2,512,256]
  float* o_prop_loc  = out + 524288;    // [2,128,2]
  float* o_prop_conf = out + 524800;    // [2,128,21]
  float* o_center    = out + 530176;    // [2,128,1]
  float* o_slp       = out + 530432;    // [2,128,512]
  float* o_elp       = out + 661504;
  float* o_scp       = out + 792576;
  float* o_ecp       = out + 923648;

  // start / end transposes of frame-level features
  {
    int total = BATCH * 512 * 256;
    transpose_slice<<<(total + 255) / 256, 256, 0, stream>>>(flf, o_start, 512, 512, 0, 256);
    transpose_slice<<<(total + 255) / 256, 256, 0, stream>>>(flf, o_end,   512, 512, 256, 256);
  }

  // ---- loc branch ----
  run_branch(stream, loc, loc_feat, flf, segments, frame_seg,
             y1024, feat2, y512, pool512, cat, propO);
  head_gemm<<<(BATCH * NLVL * 2 + 255) / 256, 256, 0, stream>>>(
      loc_head_w, loc_head_b, propO, o_prop_loc, 512, NLVL, 2);
  center_conv<<<(BATCH * NLVL + 255) / 256, 256, 0, stream>>>(
      center_w, center_b, propO, o_center, 512, NLVL);
  {
    int total = BATCH * NLVL * 512;
    transpose_slice<<<(total + 255) / 256, 256, 0, stream>>>(feat2, o_slp, 1024, NLVL, 0, 512);
    transpose_slice<<<(total + 255) / 256, 256, 0, stream>>>(feat2, o_elp, 1024, NLVL, 512, 512);
  }

  // ---- conf branch (workspace reuse is safe: same stream, in-order) ----
  run_branch(stream, conf, conf_feat, flf, segments, frame_seg,
             y1024, feat2, y512, pool512, cat, propO);
  head_gemm<<<(BATCH * NLVL * 21 + 255) / 256, 256, 0, stream>>>(
      conf_head_w, conf_head_b, propO, o_prop_conf, 512, NLVL, 21);
  {
    int total = BATCH * NLVL * 512;
    transpose_slice<<<(total + 255) / 256, 256, 0, stream>>>(feat2, o_scp, 1024, NLVL, 0, 512);
    transpose_slice<<<(total + 255) / 256, 256, 0, stream>>>(feat2, o_ecp, 1024, NLVL, 512, 512);
  }
}